// ZBLBasis_28939489641137
// MI455X (gfx1250) — compile-verified
//
#include <hip/hip_runtime.h>

// ---------------------------------------------------------------------------
// ZBL pairwise repulsion + segment-sum for MI455X (gfx1250, wave32)
//
//   - Per-edge coefficients depend only on (elem_u, elem_v): 10 elems -> 100
//     pairs. Precompute {0.5*14.3996*Zu*Zv, 1/a, 1/rmax} into a table staged
//     in LDS via the CDNA5 async global->LDS copy (ASYNCcnt tracked).
//   - Edge kernel: one pass, 4 edges/thread, b128 streaming loads (~38 MB
//     total -> ~1.7 us HBM floor), u8 per-node elem gathers (100 KB, L2/L0
//     resident), 4x v_exp_f32 + FMAs, no-return global_atomic_add_f32
//     scatter into the 400 KB L2-resident output.
// ---------------------------------------------------------------------------

#define LOG2E_F 1.44269504088896340736f

__constant__ float c_cov[119] = {
    0.2f, 0.31f, 0.28f, 1.28f, 0.96f, 0.84f, 0.76f, 0.71f, 0.66f, 0.57f, 0.58f, 1.66f,
    1.41f, 1.21f, 1.11f, 1.07f, 1.05f, 1.02f, 1.06f, 2.03f, 1.76f, 1.70f, 1.60f, 1.53f,
    1.39f, 1.39f, 1.32f, 1.26f, 1.24f, 1.32f, 1.22f, 1.22f, 1.20f, 1.19f, 1.20f, 1.20f,
    1.16f, 2.20f, 1.95f, 1.90f, 1.75f, 1.64f, 1.54f, 1.47f, 1.46f, 1.42f, 1.39f, 1.45f,
    1.44f, 1.42f, 1.39f, 1.39f, 1.38f, 1.39f, 1.40f, 2.44f, 2.15f, 2.07f, 2.04f, 2.03f,
    2.01f, 1.99f, 1.98f, 1.98f, 1.96f, 1.94f, 1.92f, 1.92f, 1.89f, 1.90f, 1.87f, 1.87f,
    1.75f, 1.70f, 1.62f, 1.51f, 1.44f, 1.41f, 1.36f, 1.36f, 1.32f, 1.45f, 1.46f, 1.48f,
    1.40f, 1.50f, 1.50f, 2.60f, 2.21f, 2.15f, 2.06f, 2.00f, 1.96f, 1.90f, 1.87f, 1.80f,
    1.69f, 0.2f, 0.2f, 0.2f, 0.2f, 0.2f, 0.2f, 0.2f, 0.2f, 0.2f, 0.2f, 0.2f,
    0.2f, 0.2f, 0.2f, 0.2f, 0.2f, 0.2f, 0.2f, 0.2f, 0.2f, 0.2f, 0.2f};

typedef int v4i __attribute__((vector_size(16)));
typedef __attribute__((address_space(1))) v4i g_v4i;  // global
typedef __attribute__((address_space(3))) v4i l_v4i;  // LDS

#if defined(__has_builtin)
#if __has_builtin(__builtin_amdgcn_global_load_async_to_lds_b128)
#define ZBL_HAVE_ASYNC_LDS 1
#endif
#endif

// -------------------------------------------------------------- zero output
__global__ __launch_bounds__(256) void zbl_zero(float* __restrict__ out, int n) {
    int i = blockIdx.x * blockDim.x + threadIdx.x;
    if (i < n) out[i] = 0.0f;
}

// ------------------------------------- per-node one-hot argmax -> elem (u8)
__global__ __launch_bounds__(256) void zbl_node_elem(
    const float* __restrict__ attrs, unsigned char* __restrict__ elem,
    int n_nodes, int n_elems) {
    int i = blockIdx.x * blockDim.x + threadIdx.x;
    if (i >= n_nodes) return;
    const float* a = attrs + (size_t)i * n_elems;
    int best = 0;
    float bv = a[0];
    for (int e = 1; e < n_elems; ++e) {
        float v = a[e];
        if (v > bv) { bv = v; best = e; }  // first-max, matches jnp.argmax
    }
    elem[i] = (unsigned char)best;
}

// --------------------------- 100-entry (elem_u, elem_v) coefficient table
// table[t] = { 0.5*14.3996*Zu*Zv, 1/a, 1/rmax, 0 }
__global__ __launch_bounds__(128) void zbl_pair_table(
    const int* __restrict__ Zs, float4* __restrict__ table, int n_elems) {
    int t = threadIdx.x;
    int npair = n_elems * n_elems;
    if (t >= npair) return;
    int eu = t / n_elems;
    int ev = t - eu * n_elems;
    int Zui = Zs[eu], Zvi = Zs[ev];
    float Zu = (float)Zui, Zv = (float)Zvi;
    float su = powf(Zu, 0.3f);
    float sv = powf(Zv, 0.3f);
    float inv_a = (su + sv) / (0.4543f * 0.529f);  // r_over_a = x * inv_a
    float rmax = c_cov[Zui] + c_cov[Zvi];
    float4 r;
    r.x = 7.1998f * Zu * Zv;  // 0.5 * 14.3996 * Zu * Zv
    r.y = inv_a;
    r.z = 1.0f / rmax;
    r.w = 0.0f;
    table[t] = r;
}

// ------------------------------------------------------ per-edge evaluation
__device__ __forceinline__ void zbl_one_edge(
    float xv, int sn, int rn,
    const unsigned char* __restrict__ node_elem,
    const float4* __restrict__ s_tab, int n_elems,
    float* __restrict__ out) {
    int eu = node_elem[sn];
    int ev = node_elem[rn];
    float4 pt = s_tab[eu * n_elems + ev];

    // phi = sum_i C_i * exp(-k_i * x / a)   via exp2 (v_exp_f32)
    float b = -(xv * pt.y) * LOG2E_F;
    float phi = 0.1818f  * exp2f(3.2f    * b)
              + 0.5099f  * exp2f(0.9423f * b)
              + 0.2802f  * exp2f(0.4028f * b)
              + 0.02817f * exp2f(0.2016f * b);

    // polynomial cutoff envelope, p = 6: 1 - 28 r^6 + 48 r^7 - 21 r^8
    float rr = xv * pt.z;
    float r2 = rr * rr;
    float r3 = r2 * rr;
    float r6 = r3 * r3;
    float env = 1.0f - 28.0f * r6 + 48.0f * r6 * rr - 21.0f * r6 * r2;
    env = (rr < 1.0f) ? env : 0.0f;

    float v = pt.x * phi * env / xv;
    // device-scope relaxed fadd -> no-return global_atomic_add_f32
    __hip_atomic_fetch_add(out + rn, v, __ATOMIC_RELAXED,
                           __HIP_MEMORY_SCOPE_AGENT);
}

// -------------------------------------------------------------- edge kernel
__global__ __launch_bounds__(256) void zbl_edges(
    const float* __restrict__ x,
    const int* __restrict__ sender,
    const int* __restrict__ receiver,
    const unsigned char* __restrict__ node_elem,
    const float4* __restrict__ pair_tab,
    float* __restrict__ out,
    int n_edges, int n_elems) {
    __shared__ float4 s_tab[256];  // npair <= 100 for this problem
    int npair = n_elems * n_elems;

#ifdef ZBL_HAVE_ASYNC_LDS
    // CDNA5 async global->LDS broadcast of the pair table (ASYNCcnt tracked)
    if ((int)threadIdx.x < npair) {
        g_v4i* g = (g_v4i*)(pair_tab + threadIdx.x);
        l_v4i* l = (l_v4i*)(&s_tab[threadIdx.x]);
        __builtin_amdgcn_global_load_async_to_lds_b128(g, l, 0, 0);
    }
    asm volatile("s_wait_asynccnt 0x0" ::: "memory");
#else
    if ((int)threadIdx.x < npair) s_tab[threadIdx.x] = pair_tab[threadIdx.x];
#endif
    __syncthreads();

    int gid = blockIdx.x * blockDim.x + threadIdx.x;
    int n_vec = n_edges >> 2;  // groups of 4 edges

    if (gid < n_vec) {
        int e0 = gid << 2;
        float4 x4 = *(const float4*)(x + e0);        // global_load_b128
        int4   s4 = *(const int4*)(sender + e0);     // global_load_b128
        int4   r4 = *(const int4*)(receiver + e0);   // global_load_b128
        zbl_one_edge(x4.x, s4.x, r4.x, node_elem, s_tab, n_elems, out);
        zbl_one_edge(x4.y, s4.y, r4.y, node_elem, s_tab, n_elems, out);
        zbl_one_edge(x4.z, s4.z, r4.z, node_elem, s_tab, n_elems, out);
        zbl_one_edge(x4.w, s4.w, r4.w, node_elem, s_tab, n_elems, out);
    }

    // scalar tail for n_edges % 4
    int tail_start = n_vec << 2;
    int te = tail_start + gid;
    if (gid < (n_edges - tail_start)) {
        zbl_one_edge(x[te], sender[te], receiver[te],
                     node_elem, s_tab, n_elems, out);
    }
}

// --------------------------------------------------------------------------
extern "C" void kernel_launch(void* const* d_in, const int* in_sizes, int n_in,
                              void* d_out, int out_size, void* d_ws, size_t ws_size,
                              hipStream_t stream) {
    const float* x              = (const float*)d_in[0];
    const float* node_attrs     = (const float*)d_in[1];
    const int*   edge_index     = (const int*)d_in[2];
    const int*   atomic_numbers = (const int*)d_in[3];

    int n_edges = in_sizes[0];      // x is [E,1]
    int n_elems = in_sizes[3];      // 10
    int n_nodes = out_size;         // 100000

    const int* sender   = edge_index;
    const int* receiver = edge_index + n_edges;
    float* out = (float*)d_out;

    // workspace layout: [ node_elem : n_nodes u8 ][ align16 ][ pair table ]
    unsigned char* node_elem = (unsigned char*)d_ws;
    size_t tab_off = ((size_t)n_nodes + 15) & ~(size_t)15;
    float4* pair_tab = (float4*)((char*)d_ws + tab_off);

    const int B = 256;
    int nblk_nodes = (n_nodes + B - 1) / B;

    zbl_zero<<<nblk_nodes, B, 0, stream>>>(out, n_nodes);
    zbl_node_elem<<<nblk_nodes, B, 0, stream>>>(node_attrs, node_elem,
                                                n_nodes, n_elems);
    zbl_pair_table<<<1, 128, 0, stream>>>(atomic_numbers, pair_tab, n_elems);

    // one thread per 4 edges, single pass
    int n_vec = n_edges >> 2;
    int work = (n_vec > 0) ? n_vec : (n_edges > 0 ? n_edges : 1);
    int nblk_edges = (work + B - 1) / B;
    zbl_edges<<<nblk_edges, B, 0, stream>>>(x, sender, receiver, node_elem,
                                            pair_tab, out, n_edges, n_elems);
}